// OrderMatching_35107062677773
// MI455X (gfx1250) — compile-verified
//
#include <hip/hip_runtime.h>
#include <hip/hip_bf16.h>

typedef __attribute__((ext_vector_type(2))) float v2f;
typedef __attribute__((ext_vector_type(8))) float v8f;

#define BATCH 32
#define NQ    32
#define ND    256
#define NL    8192

// dynamic extract from v8f via cndmask chain (no scratch spill)
__device__ __forceinline__ float sel8(v8f c, int r) {
    float x = c[0];
    x = (r == 1) ? c[1] : x;
    x = (r == 2) ? c[2] : x;
    x = (r == 3) ? c[3] : x;
    x = (r == 4) ? c[4] : x;
    x = (r == 5) ? c[5] : x;
    x = (r == 6) ? c[6] : x;
    x = (r == 7) ? c[7] : x;
    return x;
}

// ---------------------------------------------------------------------------
// Kernel 1: sim = softmax_q( (Q . T^T) * D^-1/2 ), accumulate sum of
// (1 - sim[idx(l), l]) per block. Exact-f32 path via V_WMMA_F32_16X16X4_F32.
// grid = (BATCH, NL/512), block = 256 (8 waves); each wave: 4 tiles of
// 32(q) x 16(l), K=256 in steps of 4.
// ---------------------------------------------------------------------------
__global__ __launch_bounds__(256) void
om_sim_kernel(const float* __restrict__ query, const float* __restrict__ text,
              const int* __restrict__ labels, float* __restrict__ partials) {
    const int b    = blockIdx.x;
    const int wave = threadIdx.x >> 5;
    const int lane = threadIdx.x & 31;
    const int half = lane >> 4;   // which 16-lane half
    const int m    = lane & 15;   // row (A) / column (B) index within tile

    const float* qb = query + (size_t)b * NQ * ND;
    const float* tb = text  + (size_t)b * NL * ND;
    const int*   lb = labels + (size_t)b * NL;

    const float scale = 0.0625f;  // D^-1/2 = 1/16
    float partial = 0.0f;

    const int colBase = blockIdx.y * 512 + wave * 64;

    // A-fragment base pointers: lane needs query[m][k0 + 2*half + {0,1}]
    const float* aptr0 = qb + (size_t)m * ND + 2 * half;
    const float* aptr1 = aptr0 + 16 * ND;

#pragma unroll 1
    for (int t = 0; t < 4; ++t) {
        const int col0 = colBase + t * 16;
        const float* bptr = tb + (size_t)(col0 + m) * ND + 2 * half;

        v8f acc0 = {};  // rows q = 0..15
        v8f acc1 = {};  // rows q = 16..31
#pragma unroll 4
        for (int kk = 0; kk < 64; ++kk) {
            v2f a0 = *(const v2f*)(aptr0 + kk * 4);
            v2f a1 = *(const v2f*)(aptr1 + kk * 4);
            v2f bf = __builtin_nontemporal_load((const v2f*)(bptr + kk * 4));
            acc0 = __builtin_amdgcn_wmma_f32_16x16x4_f32(false, a0, false, bf,
                                                         (short)0, acc0, false, false);
            acc1 = __builtin_amdgcn_wmma_f32_16x16x4_f32(false, a1, false, bf,
                                                         (short)0, acc1, false, false);
        }

        // ---- softmax over the 32 q-values of column n = m ----
        // lane holds 16 of them (8 per accumulator); partner lane = lane^16
        float lm = acc0[0];
#pragma unroll
        for (int r = 1; r < 8; ++r) lm = fmaxf(lm, acc0[r]);
#pragma unroll
        for (int r = 0; r < 8; ++r) lm = fmaxf(lm, acc1[r]);
        lm = fmaxf(lm, __shfl_xor(lm, 16, 32));
        const float mS = lm * scale;

        float s = 0.0f;
#pragma unroll
        for (int r = 0; r < 8; ++r) s += __expf(acc0[r] * scale - mS);
#pragma unroll
        for (int r = 0; r < 8; ++r) s += __expf(acc1[r] * scale - mS);
        s += __shfl_xor(s, 16, 32);

        // ---- picked = softmax value at row idx for this column ----
        const int lab = lb[col0 + m];
        const int idx = (lab == -1) ? (NQ - 1) : (lab - 1);  // in [0,31]
        const int r8  = idx & 7;
        const float v0 = sel8(acc0, r8);
        const float v1 = sel8(acc1, r8);
        const float v  = (idx < 16) ? v0 : v1;
        const int needHalf = (idx >> 3) & 1;  // which lane-half holds row idx
        float cand = (needHalf == half) ? __expf(v * scale - mS) : 0.0f;
        cand += __shfl_xor(cand, 16, 32);     // both halves now hold exp(picked)
        if (half == 0) partial += 1.0f - cand / s;
    }

    // wave reduce (non-zero only in lanes 0..15)
    partial += __shfl_xor(partial, 8, 32);
    partial += __shfl_xor(partial, 4, 32);
    partial += __shfl_xor(partial, 2, 32);
    partial += __shfl_xor(partial, 1, 32);

    __shared__ float red[8];
    if (lane == 0) red[wave] = partial;
    __syncthreads();
    if (threadIdx.x == 0) {
        float sum = 0.0f;
#pragma unroll
        for (int w = 0; w < 8; ++w) sum += red[w];
        partials[blockIdx.x * gridDim.y + blockIdx.y] = sum;
    }
}

// ---------------------------------------------------------------------------
// Kernel 2: Gram matrix G = Q Q^T per batch via WMMA f32 (A and B fragments
// of query load identically for the 16x16x4 layout, so the same registers
// serve both operands). One wave per batch.
// ---------------------------------------------------------------------------
__global__ __launch_bounds__(32) void
om_gram_kernel(const float* __restrict__ query, float* __restrict__ offp,
               float* __restrict__ diagp) {
    const int b    = blockIdx.x;
    const int lane = threadIdx.x;
    const int half = lane >> 4;
    const int m    = lane & 15;

    const float* qb = query + (size_t)b * NQ * ND;
    const float* pA = qb + (size_t)m * ND + 2 * half;  // rows 0..15
    const float* pB = pA + 16 * ND;                    // rows 16..31

    v8f g00 = {}, g01 = {}, g10 = {}, g11 = {};
#pragma unroll 4
    for (int kk = 0; kk < 64; ++kk) {
        v2f fA = *(const v2f*)(pA + kk * 4);
        v2f fB = *(const v2f*)(pB + kk * 4);
        g00 = __builtin_amdgcn_wmma_f32_16x16x4_f32(false, fA, false, fA, (short)0, g00, false, false);
        g01 = __builtin_amdgcn_wmma_f32_16x16x4_f32(false, fA, false, fB, (short)0, g01, false, false);
        g10 = __builtin_amdgcn_wmma_f32_16x16x4_f32(false, fB, false, fA, (short)0, g10, false, false);
        g11 = __builtin_amdgcn_wmma_f32_16x16x4_f32(false, fB, false, fB, (short)0, g11, false, false);
    }

    // lane holds G[i0 + {0,16}][j0 + {0,16}] with i0 = r + 8*half, j0 = m
    float offs = 0.0f, diag = 0.0f;
#pragma unroll
    for (int r = 0; r < 8; ++r) {
        const int i0 = r + 8 * half;
        const int j0 = m;
        float v;
        v = g00[r];
        if (i0 == j0) diag += fmaxf(1.0f - v, 0.0f); else offs += fabsf(v);
        v = g01[r];  // (i0, j0+16): never diagonal
        offs += fabsf(v);
        v = g10[r];  // (i0+16, j0): never diagonal
        offs += fabsf(v);
        v = g11[r];
        if (i0 == j0) diag += fmaxf(1.0f - v, 0.0f); else offs += fabsf(v);
    }

#pragma unroll
    for (int d = 16; d >= 1; d >>= 1) {
        offs += __shfl_xor(offs, d, 32);
        diag += __shfl_xor(diag, d, 32);
    }
    if (lane == 0) {
        offp[b]  = offs;  // sum_{i != j} |G_ij|  (each unordered pair twice)
        diagp[b] = diag;  // sum_i relu(1 - G_ii)
    }
}

// ---------------------------------------------------------------------------
// Kernel 3: deterministic fixed-order final reduction + loss assembly.
// ---------------------------------------------------------------------------
__global__ void om_finalize_kernel(const float* __restrict__ partials,
                                   const float* __restrict__ offp,
                                   const float* __restrict__ diagp,
                                   float* __restrict__ out) {
    if (threadIdx.x == 0 && blockIdx.x == 0) {
        float c = 0.0f;
        for (int i = 0; i < 512; ++i) c += partials[i];
        float ob = 0.0f, dg = 0.0f;
        for (int b = 0; b < BATCH; ++b) { ob += offp[b]; dg += diagp[b]; }
        const float contrastive = c / (float)(BATCH * NL);
        // reference: off = 0.5 * sum_{i!=j}|G|; orth = mean_b(off / 496)
        const float orth = (0.5f * ob / (float)BATCH) / 496.0f;
        const float sz   = (dg / (float)BATCH) / 31.0f;
        out[0] = contrastive + 0.5f * sz + 0.5f * orth;
    }
}

extern "C" void kernel_launch(void* const* d_in, const int* in_sizes, int n_in,
                              void* d_out, int out_size, void* d_ws, size_t ws_size,
                              hipStream_t stream) {
    const float* query  = (const float*)d_in[0];
    const float* text   = (const float*)d_in[1];
    const int*   labels = (const int*)d_in[2];
    // d_in[3] = object_num (always Q) — unused

    float* ws       = (float*)d_ws;
    float* partials = ws;        // 512 block partials
    float* offp     = ws + 512;  // 32 per-batch off-diagonal sums
    float* diagp    = ws + 544;  // 32 per-batch diag relu sums

    dim3 grid1(BATCH, NL / 512);
    om_sim_kernel<<<grid1, 256, 0, stream>>>(query, text, labels, partials);
    om_gram_kernel<<<BATCH, 32, 0, stream>>>(query, offp, diagp);
    om_finalize_kernel<<<1, 1, 0, stream>>>(partials, offp, diagp, (float*)d_out);
}